// KmeansDatasetImputation_1700807049556
// MI455X (gfx1250) — compile-verified
//
#include <hip/hip_runtime.h>
#include <hip/hip_bf16.h>
#include <stdint.h>

typedef __attribute__((ext_vector_type(2))) float v2f;
typedef __attribute__((ext_vector_type(4))) float v4f;
typedef __attribute__((ext_vector_type(8))) float v8f;

#define DIM    784
#define DIM2   1568            // augmented reduction dim
#define NCTR   512
#define NB     8192
#define NITER  392             // DIM2 / 4 k-steps
#define NPAIR  196             // k-step pairs (2 steps packed per float4)
#define TSTRIDE (NPAIR * 128)  // floats between tile blocks in bpw (= 25088 -> 100352 B)
#define LDSROW 788             // padded LDS row stride (floats); 788%64=20 -> conflict-free
#define XBUFF  (16 * LDSROW)   // floats per staged buffer (50432 B)
// dynamic LDS layout (floats): x[0,12608) m[12608,25216) redV[25216,25344) redI[25344,25472)
#define LDS_FLOATS (2 * XBUFF + 256)
#define LDS_BYTES  (LDS_FLOATS * 4)

// ---------------------------------------------------------------------------
// Async global->LDS copy (CDNA5): per-lane 16B, tracked by ASYNCcnt.
// ---------------------------------------------------------------------------
__device__ __forceinline__ void async_cp16(unsigned lds_off, const float* g) {
    asm volatile("global_load_async_to_lds_b128 %0, %1, off"
                 :: "v"(lds_off), "v"(g) : "memory");
}
__device__ __forceinline__ void wait_async0() {
    asm volatile("s_wait_asynccnt 0" ::: "memory");
}

// ---------------------------------------------------------------------------
// Stage 0: augmented centers in *pair-packed* WMMA B-fragment order.
// float4 at ((tile*196 + p)*32 + lane) = lane's fragments for k-steps 2p, 2p+1.
// Inner loop: one global_load_b128 per tile per 2 steps, fully coalesced.
// ---------------------------------------------------------------------------
__global__ __launch_bounds__(256) void build_bpw(const float* __restrict__ centers,
                                                 float* __restrict__ bpw) {
    int g = blockIdx.x * blockDim.x + threadIdx.x;     // over 32*196*32
    if (g >= 32 * NPAIR * 32) return;
    int lane = g & 31;
    int rest = g >> 5;
    int p    = rest % NPAIR;
    int tile = rest / NPAIR;
    int k   = tile * 16 + (lane & 15);
    int h2  = (lane >> 4) << 1;
    v4f val;
#pragma unroll
    for (int s = 0; s < 2; ++s) {                      // k-steps 2p, 2p+1
        int j = (2 * p + s) * 4 + h2;                  // augmented index pair (j, j+1)
        float e0, e1;
        if (j < DIM) {                                 // B'[k, 0:784] = -2*c
            e0 = -2.0f * centers[(size_t)k * DIM + j];
            e1 = -2.0f * centers[(size_t)k * DIM + j + 1];
        } else {                                       // B'[k, 784:1568] = c^2
            float c0 = centers[(size_t)k * DIM + j - DIM];
            float c1 = centers[(size_t)k * DIM + j - DIM + 1];
            e0 = c0 * c0; e1 = c1 * c1;
        }
        val[2 * s]     = e0;
        val[2 * s + 1] = e1;
    }
    *(v4f*)(bpw + (size_t)g * 4) = val;
}

// ---------------------------------------------------------------------------
// Stage 1: fused masked-distance GEMM (fp32 WMMA 16x16x4) + argmin.
// WG = 16 B-rows; 8 waves x 4 center tiles (4 accumulators each) = 512 centers.
// A tile staged once per WG into LDS (async); B streams via one base address
// + immediate tile offsets, double-buffered one pair ahead.
// ---------------------------------------------------------------------------
__global__ __launch_bounds__(256) void assign_kernel(const float* __restrict__ data,
                                                     const float* __restrict__ mask,
                                                     const float* __restrict__ bpw,
                                                     const int*   __restrict__ per_cluster_index,
                                                     int*         __restrict__ out_idx) {
    extern __shared__ float lds[];
    const int tid  = threadIdx.x;
    const int wave = tid >> 5;
    const int lane = tid & 31;
    const int half = lane >> 4;              // 0: K pair {0,1}, 1: K pair {2,3}
    const int l16  = lane & 15;
    const int off  = half * 2;
    const int rowBase = blockIdx.x * 16;

    // ---- stage x and m tiles into LDS (async, 16B chunks, padded rows) ----
    const unsigned dynbase = __builtin_amdgcn_groupstaticsize();  // all-dynamic LDS
    const unsigned xbase = dynbase;
    const unsigned mbase = dynbase + XBUFF * 4;
    for (int c = tid; c < 16 * (DIM / 4); c += 256) {
        int r = c / (DIM / 4);
        int q = c - r * (DIM / 4);
        unsigned lo = (unsigned)((r * LDSROW + q * 4) * 4);
        async_cp16(xbase + lo, data + (size_t)(rowBase + r) * DIM + q * 4);
        async_cp16(mbase + lo, mask + (size_t)(rowBase + r) * DIM + q * 4);
    }
    wait_async0();
    __syncthreads();

    const float* xs = lds;
    const float* ms = lds + XBUFF;
    const int arow = l16 * LDSROW;

    // single advancing base; 4 tiles resolved by immediate offsets t*TSTRIDE
    const float* pb = bpw + (size_t)(wave * 4) * TSTRIDE + lane * 4;

    v8f acc[4] = {};
    v4f bcur[4], bnxt[4];
#pragma unroll
    for (int t = 0; t < 4; ++t) bcur[t] = *(const v4f*)(pb + t * TSTRIDE);   // p = 0

    // Phase 1: k-steps [0,196): A' = m*x   (b prefetch one pair ahead)
    for (int p = 0; p < 98; ++p) {
        const int j = p * 8 + off;
        v2f x0 = *(const v2f*)(xs + arow + j);
        v2f m0 = *(const v2f*)(ms + arow + j);
        v2f x1 = *(const v2f*)(xs + arow + j + 4);
        v2f m1 = *(const v2f*)(ms + arow + j + 4);
        v2f a0 = x0 * m0;
        v2f a1 = x1 * m1;
#pragma unroll
        for (int t = 0; t < 4; ++t)
            bnxt[t] = *(const v4f*)(pb + (size_t)(p + 1) * 128 + t * TSTRIDE);
#pragma unroll
        for (int t = 0; t < 4; ++t) {
            v2f blo = __builtin_shufflevector(bcur[t], bcur[t], 0, 1);
            acc[t] = __builtin_amdgcn_wmma_f32_16x16x4_f32(
                false, a0, false, blo, (short)0, acc[t], false, false);
        }
#pragma unroll
        for (int t = 0; t < 4; ++t) {
            v2f bhi = __builtin_shufflevector(bcur[t], bcur[t], 2, 3);
            acc[t] = __builtin_amdgcn_wmma_f32_16x16x4_f32(
                false, a1, false, bhi, (short)0, acc[t], false, false);
        }
#pragma unroll
        for (int t = 0; t < 4; ++t) bcur[t] = bnxt[t];
    }
    // Phase 2: k-steps [196,392): A' = m  (bcur carried across the boundary)
    for (int p = 98; p < NPAIR; ++p) {
        const int j = (p - 98) * 8 + off;
        v2f a0 = *(const v2f*)(ms + arow + j);
        v2f a1 = *(const v2f*)(ms + arow + j + 4);
        const int pn = (p + 1 < NPAIR) ? (p + 1) : p;          // clamp last prefetch
#pragma unroll
        for (int t = 0; t < 4; ++t)
            bnxt[t] = *(const v4f*)(pb + (size_t)pn * 128 + t * TSTRIDE);
#pragma unroll
        for (int t = 0; t < 4; ++t) {
            v2f blo = __builtin_shufflevector(bcur[t], bcur[t], 0, 1);
            acc[t] = __builtin_amdgcn_wmma_f32_16x16x4_f32(
                false, a0, false, blo, (short)0, acc[t], false, false);
        }
#pragma unroll
        for (int t = 0; t < 4; ++t) {
            v2f bhi = __builtin_shufflevector(bcur[t], bcur[t], 2, 3);
            acc[t] = __builtin_amdgcn_wmma_f32_16x16x4_f32(
                false, a1, false, bhi, (short)0, acc[t], false, false);
        }
#pragma unroll
        for (int t = 0; t < 4; ++t) bcur[t] = bnxt[t];
    }

    // ---- argmin. D layout: VGPR v = row (v + 8*half), col = tile_base + l16 ----
    float bestV[8];
    int   bestI[8];
#pragma unroll
    for (int v = 0; v < 8; ++v) {
        bestV[v] = acc[0][v];
        bestI[v] = wave * 64 + l16;
    }
#pragma unroll
    for (int t = 1; t < 4; ++t) {
        const int col = wave * 64 + t * 16 + l16;
#pragma unroll
        for (int v = 0; v < 8; ++v) {
            float d = acc[t][v];
            if (d < bestV[v]) { bestV[v] = d; bestI[v] = col; }
        }
    }
    // cross-lane min within each 16-lane half (xor masks 1..8 never cross bit 4)
#pragma unroll
    for (int m = 8; m >= 1; m >>= 1) {
#pragma unroll
        for (int v = 0; v < 8; ++v) {
            float oV = __shfl_xor(bestV[v], m, 32);
            int   oI = __shfl_xor(bestI[v], m, 32);
            if (oV < bestV[v] || (oV == bestV[v] && oI < bestI[v])) {
                bestV[v] = oV; bestI[v] = oI;
            }
        }
    }

    float* rV = lds + 2 * XBUFF;           // [8][16]
    int*   rI = (int*)(lds + 2 * XBUFF + 128);
    if (l16 == 0) {                        // lanes 0 (rows 0..7) and 16 (rows 8..15)
#pragma unroll
        for (int v = 0; v < 8; ++v) {
            rV[wave * 16 + v + 8 * half] = bestV[v];
            rI[wave * 16 + v + 8 * half] = bestI[v];
        }
    }
    __syncthreads();

    if (tid < 16) {                        // one thread per B-row: reduce 8 waves
        float bv = rV[tid];
        int   bi = rI[tid];
#pragma unroll
        for (int w = 1; w < 8; ++w) {
            float ov = rV[w * 16 + tid];
            int   oi = rI[w * 16 + tid];
            if (ov < bv || (ov == bv && oi < bi)) { bv = ov; bi = oi; }
        }
        out_idx[rowBase + tid] = per_cluster_index[bi];
    }
}

// ---------------------------------------------------------------------------
// Stage 2: bandwidth-bound row gather, float4 (rows are 16B aligned: 784*4B)
// ---------------------------------------------------------------------------
__global__ __launch_bounds__(64) void gather_kernel(const float* __restrict__ bank,
                                                    const int*   __restrict__ idx,
                                                    float*       __restrict__ out) {
    const int b   = blockIdx.x;
    const int src = idx[b];
    const float4* s = (const float4*)(bank + (size_t)src * DIM);
    float4*       d = (float4*)(out + (size_t)b * DIM);
    for (int j = threadIdx.x; j < DIM / 4; j += 64)
        d[j] = s[j];
}

extern "C" void kernel_launch(void* const* d_in, const int* in_sizes, int n_in,
                              void* d_out, int out_size, void* d_ws, size_t ws_size,
                              hipStream_t stream) {
    const float* data    = (const float*)d_in[0];   // [8192,1,28,28]
    const float* mask    = (const float*)d_in[1];   // [8192,1,28,28]
    const float* centers = (const float*)d_in[2];   // [512,784]
    const float* bank    = (const float*)d_in[3];   // [50000,784]
    const int*   pci     = (const int*)d_in[4];     // [512]
    float* out = (float*)d_out;

    float* bpw  = (float*)d_ws;                                   // 3.2 MB fragment-order B'
    int*   widx = (int*)((char*)d_ws + (size_t)32 * TSTRIDE * sizeof(float));

    build_bpw<<<(32 * NPAIR * 32 + 255) / 256, 256, 0, stream>>>(centers, bpw);
    assign_kernel<<<NB / 16, 256, LDS_BYTES, stream>>>(data, mask, bpw, pci, widx);
    gather_kernel<<<NB, 64, 0, stream>>>(bank, widx, out);
}